// GCN_28887950032998
// MI455X (gfx1250) — compile-verified
//
#include <hip/hip_runtime.h>
#include <hip/hip_bf16.h>

// ---------------------------------------------------------------------------
// ChebConv GCN for MI455X (gfx1250).
//  - graph normalization + CSR-by-dst build (counting sort, no runtime sorts)
//  - SpMM propagates: one workgroup per destination row, atomically-free,
//    L2-resident feature gathers with global_prefetch
//  - dense feature transforms via V_WMMA_F32_16X16X4_F32 (exact fp32 path)
//  - small FC head as dot-product kernels
// ---------------------------------------------------------------------------

#define NN 4096
#define EE 131072
#define BB 64

typedef __attribute__((ext_vector_type(2))) float v2f;
typedef __attribute__((ext_vector_type(8))) float v8f;

// ---------------------------- graph preprocessing --------------------------

__global__ __launch_bounds__(256) void init_kernel(float* deg, int* count) {
    int n = blockIdx.x * blockDim.x + threadIdx.x;
    if (n < NN) { deg[n] = 0.f; count[n] = 0; }
}

__global__ __launch_bounds__(256) void deg_kernel(const int* __restrict__ src,
                                                  const float* __restrict__ ew,
                                                  float* __restrict__ deg) {
    int e = blockIdx.x * blockDim.x + threadIdx.x;
    if (e < EE) atomicAdd(&deg[src[e]], ew[e]);
}

__global__ __launch_bounds__(256) void dinv_kernel(float* deg) {
    int n = blockIdx.x * blockDim.x + threadIdx.x;
    if (n < NN) { float d = deg[n]; deg[n] = (d > 0.f) ? rsqrtf(d) : 0.f; }
}

__global__ __launch_bounds__(256) void nw_kernel(const int* __restrict__ src,
                                                 const int* __restrict__ dst,
                                                 const float* __restrict__ ew,
                                                 const float* __restrict__ dinv,
                                                 float* __restrict__ nw,
                                                 int* __restrict__ count) {
    int e = blockIdx.x * blockDim.x + threadIdx.x;
    if (e < EE) {
        nw[e] = -dinv[src[e]] * ew[e] * dinv[dst[e]];
        atomicAdd(&count[dst[e]], 1);
    }
}

__global__ void scan_kernel(const int* __restrict__ count,
                            int* __restrict__ rowptr,
                            int* __restrict__ fillptr) {
    if (blockIdx.x == 0 && threadIdx.x == 0) {
        int run = 0;
        rowptr[0] = 0;
        for (int n = 0; n < NN; ++n) {
            fillptr[n] = run;
            run += count[n];
            rowptr[n + 1] = run;
        }
    }
}

__global__ __launch_bounds__(256) void scatter_kernel(const int* __restrict__ src,
                                                      const int* __restrict__ dst,
                                                      const float* __restrict__ nw,
                                                      int* __restrict__ fillptr,
                                                      int* __restrict__ ccol,
                                                      float* __restrict__ cval) {
    int e = blockIdx.x * blockDim.x + threadIdx.x;
    if (e < EE) {
        int p = atomicAdd(&fillptr[dst[e]], 1);
        ccol[p] = src[e];
        cval[p] = nw[e];
    }
}

// x: [B, N] -> z: [N, B]
__global__ __launch_bounds__(256) void transpose_kernel(const float* __restrict__ x,
                                                        float* __restrict__ z) {
    int idx = blockIdx.x * blockDim.x + threadIdx.x;
    if (idx < NN * BB) {
        int n = idx >> 6;       // /BB
        int b = idx & (BB - 1);
        z[idx] = x[(size_t)b * NN + n];
    }
}

// ------------------------------- SpMM --------------------------------------
// out[n, 0:F] = alpha * sum_{edges into n} val * z[col, 0:F]  (+ beta*t0[n,0:F])
template <int F, int TPB, int EPT>
__global__ __launch_bounds__(TPB)
void spmm_kernel(const int* __restrict__ rowptr, const int* __restrict__ ccol,
                 const float* __restrict__ cval, const float* __restrict__ z,
                 const float* __restrict__ t0, float* __restrict__ out,
                 float alpha, float beta) {
    const int n = blockIdx.x;
    const int t = threadIdx.x;
    float acc[EPT];
#pragma unroll
    for (int j = 0; j < EPT; ++j) acc[j] = 0.f;

    const int s = rowptr[n];
    const int e = rowptr[n + 1];
    for (int i = s; i < e; ++i) {
        const int c   = ccol[i];
        const float v = cval[i];
        const float* zr = z + (size_t)c * F + t;
        if (i + 1 < e) {
            // emits global_prefetch_b8: pull next source row toward the WGP
            __builtin_prefetch(z + (size_t)ccol[i + 1] * F + t, 0, 0);
        }
#pragma unroll
        for (int j = 0; j < EPT; ++j) acc[j] += v * zr[j * TPB];
    }

    float* o = out + (size_t)n * F + t;
    if (t0) {
        const float* tz = t0 + (size_t)n * F + t;
#pragma unroll
        for (int j = 0; j < EPT; ++j) o[j * TPB] = alpha * acc[j] + beta * tz[j * TPB];
    } else {
#pragma unroll
        for (int j = 0; j < EPT; ++j) o[j * TPB] = alpha * acc[j];
    }
}

// --------------------- dense transform: WMMA f32 GEMM ----------------------
// A: [M=262144 x 32] row-major, W: [32 x WCOLS] row-major, OUT: [M x 32].
// Each wave computes one 16x16 tile (x NTILES column tiles) with
// V_WMMA_F32_16X16X4_F32 over K=32 (8 k-steps).
template <int NTILES, int WCOLS>
__global__ __launch_bounds__(256)
void cheb_gemm_wmma(const float* __restrict__ A, const float* __restrict__ W,
                    const float* __restrict__ bias, float* __restrict__ OUT,
                    int accumulate, int fuse_bias_relu) {
    const int lane    = threadIdx.x & 31;
    const int wave    = threadIdx.x >> 5;
    const int row0    = (blockIdx.x * 8 + wave) * 16;
    const int halfsel = lane >> 4;          // 0 for lanes 0-15, 1 for 16-31
    const int kb      = halfsel * 2;        // K sub-offset per ISA A/B layout
    const int col     = lane & 15;
    const float* Arow = A + (size_t)(row0 + col) * 32;

    v8f c0 = {};
    v8f c1 = {};
#pragma unroll
    for (int ks = 0; ks < 8; ++ks) {
        const int k = ks * 4;
        v2f a;
        a.x = Arow[k + kb];
        a.y = Arow[k + kb + 1];
        v2f b0;
        b0.x = W[(k + kb) * WCOLS + col];
        b0.y = W[(k + kb + 1) * WCOLS + col];
        c0 = __builtin_amdgcn_wmma_f32_16x16x4_f32(false, a, false, b0,
                                                   (short)0, c0, false, false);
        if (NTILES == 2) {
            v2f b1;
            b1.x = W[(k + kb) * WCOLS + 16 + col];
            b1.y = W[(k + kb + 1) * WCOLS + 16 + col];
            c1 = __builtin_amdgcn_wmma_f32_16x16x4_f32(false, a, false, b1,
                                                       (short)0, c1, false, false);
        }
    }

#pragma unroll
    for (int j = 0; j < 8; ++j) {
        const int orow = row0 + j + halfsel * 8;   // C/D layout: vgpr j -> M=j / j+8
        {
            const size_t addr = (size_t)orow * 32 + col;
            float v = c0[j];
            if (accumulate) v += OUT[addr];
            if (fuse_bias_relu) v = fmaxf(v + bias[col], 0.f);
            OUT[addr] = v;
        }
        if (NTILES == 2) {
            const size_t addr = (size_t)orow * 32 + 16 + col;
            float v = c1[j];
            if (accumulate) v += OUT[addr];
            if (fuse_bias_relu) v = fmaxf(v + bias[16 + col], 0.f);
            OUT[addr] = v;
        }
    }
}

// --------------------------- layer-1 transform -----------------------------
// h[n,b,co] = relu( sum_k t_k[n,b] * w1[k,co] + b1[co] ),  Cin = 1
__global__ __launch_bounds__(256)
void l1_transform(const float* __restrict__ t0, const float* __restrict__ t1,
                  const float* __restrict__ t2, const float* __restrict__ t3,
                  const float* __restrict__ w, const float* __restrict__ bias,
                  float* __restrict__ h) {
    int idx = blockIdx.x * blockDim.x + threadIdx.x;    // over N*B*32
    if (idx >= NN * BB * 32) return;
    int co = idx & 31;
    int nb = idx >> 5;
    float v = t0[nb] * w[0 * 32 + co] + t1[nb] * w[1 * 32 + co] +
              t2[nb] * w[2 * 32 + co] + t3[nb] * w[3 * 32 + co] + bias[co];
    h[idx] = fmaxf(v, 0.f);
}

// pad w5 [4,32,4] -> [4,32,16] (zeros), b5 [4] -> [16]
__global__ __launch_bounds__(256)
void padw5_kernel(const float* __restrict__ w5, const float* __restrict__ b5,
                  float* __restrict__ wp, float* __restrict__ bp) {
    int idx = blockIdx.x * blockDim.x + threadIdx.x;
    if (idx < 4 * 32 * 16) {
        int c = idx & 15;
        int r = (idx >> 4) & 31;
        int k = idx >> 9;
        wp[idx] = (c < 4) ? w5[(k * 32 + r) * 4 + c] : 0.f;
    } else if (idx < 4 * 32 * 16 + 16) {
        int c = idx - 4 * 32 * 16;
        bp[c] = (c < 4) ? b5[c] : 0.f;
    }
}

// ------------------------------- FC head -----------------------------------
// y[b,j] = fb1[j] + sum_{n,c<4} h5[n,b,c] * fw1[n*4+c, j]
__global__ __launch_bounds__(256)
void fc1_kernel(const float* __restrict__ h5, const float* __restrict__ w,
                const float* __restrict__ bias, float* __restrict__ y) {
    int idx = blockIdx.x * blockDim.x + threadIdx.x;    // 64*128
    if (idx >= BB * 128) return;
    int j = idx & 127;
    int b = idx >> 7;
    float acc = bias[j];
    for (int n = 0; n < NN; ++n) {
        const float* hr = h5 + ((size_t)n * BB + b) * 32;
        const float* wr = w + (size_t)(n * 4) * 128 + j;
#pragma unroll
        for (int c = 0; c < 4; ++c) acc += hr[c] * wr[c * 128];
    }
    y[idx] = acc;
}

__global__ __launch_bounds__(256)
void fc2_kernel(const float* __restrict__ y1, const float* __restrict__ w,
                const float* __restrict__ bias, float* __restrict__ y2) {
    int idx = blockIdx.x * blockDim.x + threadIdx.x;    // 64*128
    if (idx >= BB * 128) return;
    int j = idx & 127;
    int b = idx >> 7;
    float acc = bias[j];
    for (int i = 0; i < 128; ++i) acc += y1[b * 128 + i] * w[i * 128 + j];
    y2[idx] = acc;
}

__global__ __launch_bounds__(64)
void fc3_kernel(const float* __restrict__ y2, const float* __restrict__ w,
                const float* __restrict__ bias, float* __restrict__ out) {
    int idx = blockIdx.x * blockDim.x + threadIdx.x;    // 64*9
    if (idx >= BB * 9) return;
    int j = idx % 9;
    int b = idx / 9;
    float acc = bias[j];
    for (int i = 0; i < 128; ++i) acc += y2[b * 128 + i] * w[i * 9 + j];
    out[idx] = acc;
}

// ------------------------------- launch ------------------------------------

extern "C" void kernel_launch(void* const* d_in, const int* in_sizes, int n_in,
                              void* d_out, int out_size, void* d_ws, size_t ws_size,
                              hipStream_t stream) {
    const float* x   = (const float*)d_in[0];
    const int*   src = (const int*)d_in[1];
    const int*   dst = (const int*)d_in[2];
    const float* ew  = (const float*)d_in[3];
    const float* w1  = (const float*)d_in[4];
    const float* b1  = (const float*)d_in[5];
    const float* w2  = (const float*)d_in[6];
    const float* b2  = (const float*)d_in[7];
    const float* w3  = (const float*)d_in[8];
    const float* b3  = (const float*)d_in[9];
    const float* w4  = (const float*)d_in[10];
    const float* b4  = (const float*)d_in[11];
    const float* w5  = (const float*)d_in[12];
    const float* b5  = (const float*)d_in[13];
    const float* fw1 = (const float*)d_in[14];
    const float* fb1 = (const float*)d_in[15];
    const float* fw2 = (const float*)d_in[16];
    const float* fb2 = (const float*)d_in[17];
    const float* fw3 = (const float*)d_in[18];
    const float* fb3 = (const float*)d_in[19];
    float* out = (float*)d_out;

    char* ws = (char*)d_ws;
    size_t off = 0;
    auto alloc = [&](size_t bytes) -> void* {
        void* p = ws + off;
        off = (off + bytes + 255) & ~(size_t)255;
        return p;
    };

    float* deg     = (float*)alloc((size_t)NN * 4);
    int*   count   = (int*)alloc((size_t)NN * 4);
    int*   rowptr  = (int*)alloc((size_t)(NN + 1) * 4);
    int*   fillptr = (int*)alloc((size_t)NN * 4);
    float* nw      = (float*)alloc((size_t)EE * 4);
    int*   ccol    = (int*)alloc((size_t)EE * 4);
    float* cval    = (float*)alloc((size_t)EE * 4);
    float* t0      = (float*)alloc((size_t)NN * BB * 4);
    float* t1      = (float*)alloc((size_t)NN * BB * 4);
    float* t2      = (float*)alloc((size_t)NN * BB * 4);
    float* t3      = (float*)alloc((size_t)NN * BB * 4);
    float* wp5     = (float*)alloc((size_t)4 * 32 * 16 * 4);
    float* bp5     = (float*)alloc((size_t)16 * 4);
    float* y1      = (float*)alloc((size_t)BB * 128 * 4);
    float* y2      = (float*)alloc((size_t)BB * 128 * 4);
    float* buf[4];
    for (int i = 0; i < 4; ++i) buf[i] = (float*)alloc((size_t)NN * BB * 32 * 4);

    // ---- graph normalization + CSR build ----
    init_kernel<<<(NN + 255) / 256, 256, 0, stream>>>(deg, count);
    deg_kernel<<<EE / 256, 256, 0, stream>>>(src, ew, deg);
    dinv_kernel<<<(NN + 255) / 256, 256, 0, stream>>>(deg);
    nw_kernel<<<EE / 256, 256, 0, stream>>>(src, dst, ew, deg, nw, count);
    scan_kernel<<<1, 1, 0, stream>>>(count, rowptr, fillptr);
    scatter_kernel<<<EE / 256, 256, 0, stream>>>(src, dst, nw, fillptr, ccol, cval);

    // ---- layer 1 (Cin=1) ----
    transpose_kernel<<<(NN * BB) / 256, 256, 0, stream>>>(x, t0);
    spmm_kernel<BB, BB, 1><<<NN, BB, 0, stream>>>(rowptr, ccol, cval, t0, nullptr, t1, 1.f, 0.f);
    spmm_kernel<BB, BB, 1><<<NN, BB, 0, stream>>>(rowptr, ccol, cval, t1, t0, t2, 2.f, -1.f);
    spmm_kernel<BB, BB, 1><<<NN, BB, 0, stream>>>(rowptr, ccol, cval, t2, t1, t3, 2.f, -1.f);
    l1_transform<<<(NN * BB * 32) / 256, 256, 0, stream>>>(t0, t1, t2, t3, w1, b1, buf[0]);

    const int MB = (NN * BB) / 128;   // GEMM blocks: 128 rows per block (8 waves x 16)
    constexpr int F = BB * 32;        // 2048 features per node

    // ---- layers 2..4 (Cin=Cout=32) ----
    const float* Wl[3] = {w2, w3, w4};
    const float* Bl[3] = {b2, b3, b4};
    float* h  = buf[0];
    float* fA = buf[1];
    float* fB = buf[2];
    float* fC = buf[3];
    for (int L = 0; L < 3; ++L) {
        const float* W  = Wl[L];
        const float* Bv = Bl[L];
        float* outb = fA;
        float* T1   = fB;
        float* T2   = fC;
        float* T3   = h;   // old T0 buffer is free by the time T3 is written

        cheb_gemm_wmma<2, 32><<<MB, 256, 0, stream>>>(h, W + 0 * 1024, nullptr, outb, 0, 0);
        spmm_kernel<F, 256, 8><<<NN, 256, 0, stream>>>(rowptr, ccol, cval, h, nullptr, T1, 1.f, 0.f);
        cheb_gemm_wmma<2, 32><<<MB, 256, 0, stream>>>(T1, W + 1 * 1024, nullptr, outb, 1, 0);
        spmm_kernel<F, 256, 8><<<NN, 256, 0, stream>>>(rowptr, ccol, cval, T1, h, T2, 2.f, -1.f);
        cheb_gemm_wmma<2, 32><<<MB, 256, 0, stream>>>(T2, W + 2 * 1024, nullptr, outb, 1, 0);
        spmm_kernel<F, 256, 8><<<NN, 256, 0, stream>>>(rowptr, ccol, cval, T2, T1, T3, 2.f, -1.f);
        cheb_gemm_wmma<2, 32><<<MB, 256, 0, stream>>>(T3, W + 3 * 1024, Bv, outb, 1, 1);

        float* oldh = h;
        h  = outb;
        fA = fB;
        fB = fC;
        fC = oldh;
    }

    // ---- layer 5 (Cout=4, padded to 16) ----
    padw5_kernel<<<(4 * 32 * 16 + 16 + 255) / 256, 256, 0, stream>>>(w5, b5, wp5, bp5);
    {
        float* outb = fA;
        float* T1   = fB;
        float* T2   = fC;
        float* T3   = h;
        cheb_gemm_wmma<1, 16><<<MB, 256, 0, stream>>>(h, wp5 + 0 * 512, nullptr, outb, 0, 0);
        spmm_kernel<F, 256, 8><<<NN, 256, 0, stream>>>(rowptr, ccol, cval, h, nullptr, T1, 1.f, 0.f);
        cheb_gemm_wmma<1, 16><<<MB, 256, 0, stream>>>(T1, wp5 + 1 * 512, nullptr, outb, 1, 0);
        spmm_kernel<F, 256, 8><<<NN, 256, 0, stream>>>(rowptr, ccol, cval, T1, h, T2, 2.f, -1.f);
        cheb_gemm_wmma<1, 16><<<MB, 256, 0, stream>>>(T2, wp5 + 2 * 512, nullptr, outb, 1, 0);
        spmm_kernel<F, 256, 8><<<NN, 256, 0, stream>>>(rowptr, ccol, cval, T2, T1, T3, 2.f, -1.f);
        cheb_gemm_wmma<1, 16><<<MB, 256, 0, stream>>>(T3, wp5 + 3 * 512, bp5, outb, 1, 1);
        h = outb;   // h5: [N, B, 32] with valid cols 0..3
    }

    // ---- FC head ----
    fc1_kernel<<<(BB * 128 + 255) / 256, 256, 0, stream>>>(h, fw1, fb1, y1);
    fc2_kernel<<<(BB * 128 + 255) / 256, 256, 0, stream>>>(y1, fw2, fb2, y2);
    fc3_kernel<<<(BB * 9 + 63) / 64, 64, 0, stream>>>(y2, fw3, fb3, out);
}